// EdgeGCN_36017595744488
// MI455X (gfx1250) — compile-verified
//
#include <hip/hip_runtime.h>
#include <math.h>

#define DH 128
#define TB 256

typedef __attribute__((ext_vector_type(2))) float v2f;
typedef __attribute__((ext_vector_type(8))) float v8f;

// ---------------- degree / norm precompute ----------------

__global__ void k_init_deg(float* deg, int n) {
    int i = blockIdx.x * TB + threadIdx.x;
    if (i < n) deg[i] = 1.0f;               // self-loop weight
}

__global__ void k_deg_edges(const long long* __restrict__ ei,
                            const float* __restrict__ ew,
                            float* deg, int n) {
    int e = blockIdx.x * TB + threadIdx.x;
    if (e < n) {
        int dst = (int)ei[n + e];
        atomicAdd(&deg[dst], ew[e]);
    }
}

__global__ void k_dinv(float* dinv, int n) {
    int i = blockIdx.x * TB + threadIdx.x;
    if (i < n) {
        float d = dinv[i];
        dinv[i] = (d > 0.0f) ? rsqrtf(d) : 0.0f;
    }
}

__global__ void k_norm(const long long* __restrict__ ei,
                       const float* __restrict__ ew,
                       const float* __restrict__ dinv,
                       float* __restrict__ norm, int n) {
    int e = blockIdx.x * TB + threadIdx.x;
    if (e < n) {
        int s = (int)ei[e];
        int d = (int)ei[n + e];
        norm[e] = dinv[s] * ew[e] * dinv[d];
    }
}

// ---------------- layer 1 (x0 = ew[:,None], h1 = ew*W1) ----------------

// agg[i][j] = selfnorm_i * ew[i] * W1[j]   (full overwrite: no poison reliance)
__global__ void k_l1_init(const float* __restrict__ ew,
                          const float* __restrict__ dinv,
                          const float* __restrict__ W1,
                          float* __restrict__ agg, int n) {
    int gid = blockIdx.x * TB + threadIdx.x;
    int i = gid >> 5;
    int j = (gid & 31) * 4;
    if (i < n) {
        float c = dinv[i] * dinv[i] * ew[i];
        float4 w = *(const float4*)(W1 + j);
        float4 o = make_float4(c * w.x, c * w.y, c * w.z, c * w.w);
        *(float4*)(agg + (size_t)i * DH + j) = o;
    }
}

__global__ void k_l1_edges(const long long* __restrict__ ei,
                           const float* __restrict__ ew,
                           const float* __restrict__ norm,
                           const float* __restrict__ W1,
                           float* __restrict__ agg, int n) {
    int gid = blockIdx.x * TB + threadIdx.x;
    int e = gid >> 5;
    int j = (gid & 31) * 4;
    if (e < n) {
        int s = (int)ei[e];
        int d = (int)ei[n + e];
        float c = norm[e] * ew[s];
        float4 w = *(const float4*)(W1 + j);
        float* p = agg + (size_t)d * DH + j;
        atomicAdd(p + 0, c * w.x);
        atomicAdd(p + 1, c * w.y);
        atomicAdd(p + 2, c * w.z);
        atomicAdd(p + 3, c * w.w);
    }
}

__global__ void k_bias_relu(float* __restrict__ x,
                            const float* __restrict__ b, int n) {
    int gid = blockIdx.x * TB + threadIdx.x;
    int i = gid >> 5;
    int j = (gid & 31) * 4;
    if (i < n) {
        float4 v = *(float4*)(x + (size_t)i * DH + j);
        float4 bb = *(const float4*)(b + j);
        v.x = fmaxf(v.x + bb.x, 0.0f);
        v.y = fmaxf(v.y + bb.y, 0.0f);
        v.z = fmaxf(v.z + bb.z, 0.0f);
        v.w = fmaxf(v.w + bb.w, 0.0f);
        *(float4*)(x + (size_t)i * DH + j) = v;
    }
}

// ---------------- layer 2 GEMM: H = X @ W2 via V_WMMA_F32_16X16X4_F32 --------
// One wave owns a 16-row strip and all 8 column tiles (X streamed exactly once).
// W2 (64 KB) staged in LDS. All WMMA-executing branches are wave-uniform so
// EXEC is all-ones when the matrix op issues. Store epilogue has a wave-uniform
// fast path (full strip): 64 unguarded global_store_b32 at immediate offsets.

__global__ __launch_bounds__(TB) void k_gemm_w2(const float* __restrict__ X,
                                                const float* __restrict__ W,
                                                float* __restrict__ H, int n) {
    __shared__ float ldsW[DH * DH];         // 64 KB of the 320 KB WGP LDS
    const int tid = threadIdx.x;

    // cooperative 128-bit loads of W2 into LDS
    const float4* Wv = (const float4*)W;
    float4* Lv = (float4*)ldsW;
    for (int i = tid; i < (DH * DH) / 4; i += TB) Lv[i] = Wv[i];
    __syncthreads();

    const int wave = tid >> 5;
    const int lane = tid & 31;
    const int m    = lane & 15;             // row (A) / col (B) within tile
    const int kb   = (lane >> 4) << 1;      // K pair base: 0 or 2
    const int half = lane >> 4;

    int strip = blockIdx.x * 8 + wave;      // 16-row strip id (wave-uniform)
    int row0 = strip * 16;
    if (row0 >= n) return;                  // whole wave exits together

    v8f acc[8] = {};

    int r = row0 + m;
    if (r >= n) r = n - 1;                  // clamp tail loads (stores guarded)
    const float* Xrow = X + (size_t)r * DH;

    for (int k0 = 0; k0 < DH; k0 += 4) {
        v2f a;
        a.x = Xrow[k0 + kb];
        a.y = Xrow[k0 + kb + 1];
#pragma unroll
        for (int t = 0; t < 8; ++t) {
            v2f b;
            int col = t * 16 + m;
            b.x = ldsW[(k0 + kb) * DH + col];
            b.y = ldsW[(k0 + kb + 1) * DH + col];
            acc[t] = __builtin_amdgcn_wmma_f32_16x16x4_f32(
                false, a, false, b, (short)0, acc[t], false, false);
        }
    }

    if (row0 + 16 <= n) {
        // fast path: full strip, no per-element guards, constant-offset stores
#pragma unroll
        for (int t = 0; t < 8; ++t) {
            float* p = H + (size_t)(row0 + half * 8) * DH + t * 16 + m;
#pragma unroll
            for (int i = 0; i < 8; ++i) {
                p[i * DH] = acc[t][i];
            }
        }
    } else {
        // tail strip (only if n % 16 != 0)
#pragma unroll
        for (int t = 0; t < 8; ++t) {
            int col = t * 16 + m;
#pragma unroll
            for (int i = 0; i < 8; ++i) {
                int row = row0 + i + half * 8;
                if (row < n) H[(size_t)row * DH + col] = acc[t][i];
            }
        }
    }
}

// ---------------- layer 2 aggregation ----------------

__global__ void k_l2_init(const float* __restrict__ h,
                          const float* __restrict__ dinv,
                          float* __restrict__ agg, int n) {
    int gid = blockIdx.x * TB + threadIdx.x;
    int i = gid >> 5;
    int j = (gid & 31) * 4;
    if (i < n) {
        float c = dinv[i] * dinv[i];
        float4 v = *(const float4*)(h + (size_t)i * DH + j);
        float4 o = make_float4(c * v.x, c * v.y, c * v.z, c * v.w);
        *(float4*)(agg + (size_t)i * DH + j) = o;
    }
}

__global__ void k_l2_edges(const long long* __restrict__ ei,
                           const float* __restrict__ norm,
                           const float* __restrict__ h,
                           float* __restrict__ agg, int n) {
    int gid = blockIdx.x * TB + threadIdx.x;
    int e = gid >> 5;
    int j = (gid & 31) * 4;
    if (e < n) {
        int s = (int)ei[e];
        int d = (int)ei[n + e];
        float c = norm[e];
        float4 v = *(const float4*)(h + (size_t)s * DH + j);
        float* p = agg + (size_t)d * DH + j;
        atomicAdd(p + 0, c * v.x);
        atomicAdd(p + 1, c * v.y);
        atomicAdd(p + 2, c * v.z);
        atomicAdd(p + 3, c * v.w);
    }
}

// ---------------- layer 3 (128 -> 2) ----------------

__global__ void k_l3_h(const float* __restrict__ x,
                       const float* __restrict__ W3,
                       float* __restrict__ h3, int n) {
    int i = blockIdx.x * TB + threadIdx.x;
    if (i < n) {
        const float* xr = x + (size_t)i * DH;
        float a0 = 0.0f, a1 = 0.0f;
#pragma unroll 4
        for (int j = 0; j < DH; ++j) {
            float xv = xr[j];
            a0 += xv * W3[j * 2 + 0];
            a1 += xv * W3[j * 2 + 1];
        }
        h3[i * 2 + 0] = a0;
        h3[i * 2 + 1] = a1;
    }
}

__global__ void k_l3_init(const float* __restrict__ h3,
                          const float* __restrict__ dinv,
                          float* __restrict__ out, int n) {
    int i = blockIdx.x * TB + threadIdx.x;
    if (i < n) {
        float c = dinv[i] * dinv[i];
        out[i * 2 + 0] = c * h3[i * 2 + 0];
        out[i * 2 + 1] = c * h3[i * 2 + 1];
    }
}

__global__ void k_l3_edges(const long long* __restrict__ ei,
                           const float* __restrict__ norm,
                           const float* __restrict__ h3,
                           float* __restrict__ out, int n) {
    int e = blockIdx.x * TB + threadIdx.x;
    if (e < n) {
        int s = (int)ei[e];
        int d = (int)ei[n + e];
        float c = norm[e];
        atomicAdd(&out[d * 2 + 0], c * h3[s * 2 + 0]);
        atomicAdd(&out[d * 2 + 1], c * h3[s * 2 + 1]);
    }
}

__global__ void k_final(float* __restrict__ out,
                        const float* __restrict__ b3, int n) {
    int i = blockIdx.x * TB + threadIdx.x;
    if (i < n) {
        float y0 = fmaxf(out[i * 2 + 0] + b3[0], 0.0f);
        float y1 = fmaxf(out[i * 2 + 1] + b3[1], 0.0f);
        float mx = fmaxf(y0, y1);
        float lse = mx + logf(expf(y0 - mx) + expf(y1 - mx));
        out[i * 2 + 0] = y0 - lse;
        out[i * 2 + 1] = y1 - lse;
    }
}

// ---------------- launcher ----------------

extern "C" void kernel_launch(void* const* d_in, const int* in_sizes, int n_in,
                              void* d_out, int out_size, void* d_ws, size_t ws_size,
                              hipStream_t stream) {
    const long long* ei = (const long long*)d_in[0];   // (2, N) int64
    const float* ew = (const float*)d_in[1];
    const float* W1 = (const float*)d_in[2];
    const float* b1 = (const float*)d_in[3];
    const float* W2 = (const float*)d_in[4];
    const float* b2 = (const float*)d_in[5];
    const float* W3 = (const float*)d_in[6];
    const float* b3 = (const float*)d_in[7];
    float* out = (float*)d_out;

    const int n = in_sizes[1];                         // N = 200000

    float* ws   = (float*)d_ws;
    float* dinv = ws;                                  // N
    float* norm = ws + (size_t)n;                      // N
    float* bufA = ws + 2 * (size_t)n;                  // N*128
    float* bufB = bufA + (size_t)n * DH;               // N*128

    const int gN  = (n + TB - 1) / TB;                 // 1 thread / node-or-edge
    const int gNF = (int)(((size_t)n * 32 + TB - 1) / TB); // 32 threads / row
    const int gG  = (((n + 15) / 16) + 7) / 8;         // gemm: 8 strips / block

    // norm precompute
    k_init_deg<<<gN, TB, 0, stream>>>(dinv, n);
    k_deg_edges<<<gN, TB, 0, stream>>>(ei, ew, dinv, n);
    k_dinv<<<gN, TB, 0, stream>>>(dinv, n);
    k_norm<<<gN, TB, 0, stream>>>(ei, ew, dinv, norm, n);

    // layer 1: h1 = ew * W1 (computed on the fly), aggregate -> x1 in bufA
    k_l1_init<<<gNF, TB, 0, stream>>>(ew, dinv, W1, bufA, n);
    k_l1_edges<<<gNF, TB, 0, stream>>>(ei, ew, norm, W1, bufA, n);
    k_bias_relu<<<gNF, TB, 0, stream>>>(bufA, b1, n);

    // layer 2: h2 = x1 @ W2 (WMMA) -> bufB, aggregate -> x2 in bufA
    k_gemm_w2<<<gG, TB, 0, stream>>>(bufA, W2, bufB, n);
    k_l2_init<<<gNF, TB, 0, stream>>>(bufB, dinv, bufA, n);
    k_l2_edges<<<gNF, TB, 0, stream>>>(ei, norm, bufB, bufA, n);
    k_bias_relu<<<gNF, TB, 0, stream>>>(bufA, b2, n);

    // layer 3: h3 = x2 @ W3 -> bufB[0:2N), aggregate into d_out, finalize
    k_l3_h<<<gN, TB, 0, stream>>>(bufA, W3, bufB, n);
    k_l3_init<<<gN, TB, 0, stream>>>(bufB, dinv, out, n);
    k_l3_edges<<<gN, TB, 0, stream>>>(ei, norm, bufB, out, n);
    k_final<<<gN, TB, 0, stream>>>(out, b3, n);
}